// BlockAttnResRouter_86208583565578
// MI455X (gfx1250) — compile-verified
//
#include <hip/hip_runtime.h>
#include <hip/hip_bf16.h>
#include <stdint.h>

// ---------------------------------------------------------------------------
// Fused RMSNorm-score + softmax-router + weighted-sum kernel (BlockAttnResRouter)
//   sources: (B,S,N,D) f32,  N=6, D=1024
//   out:     h (B,S,D) f32   followed by alpha_mem (B,S) f32
// Bandwidth-bound (~470 MB @ 23.3 TB/s ~= 20us floor) -> stream sources once
// through the CDNA5 async global->LDS path (ASYNCcnt), double buffered, with
// non-temporal hints since the stream (403MB in + 67MB out) exceeds the 192MB L2.
// ---------------------------------------------------------------------------

#define N_SRC   6
#define DIM     1024
#define TPB     256            // 8 wave32
#define WAVES   (TPB / 32)
#define PPB     8              // positions per block
#define RMS_EPS 1e-6f

// Temporal hint: TH_LOAD_NT = 1 (non-temporal) in the gfx1250 CPol field.
#define CPOL_NT 1

// Typed pointers matching the gfx1250 async-copy builtin prototype
// (vV4i*1, V4i*3, Ii, Ii): global int4* source, LDS int4* destination.
typedef __attribute__((ext_vector_type(4))) int v4i_t;
typedef __attribute__((ext_vector_type(4))) float v4f_t;   // native vec for NT store
typedef __attribute__((address_space(1))) v4i_t* gas_v4i_ptr;
typedef __attribute__((address_space(3))) v4i_t* las_v4i_ptr;

// ---- CDNA5 async global->LDS b128 copy (tracked by ASYNCcnt) --------------
__device__ __forceinline__ void async_f4_to_lds(const float* gsrc, float* lds_dst) {
#if defined(__has_builtin) && __has_builtin(__builtin_amdgcn_global_load_async_to_lds_b128)
  __builtin_amdgcn_global_load_async_to_lds_b128(
      (gas_v4i_ptr)(uintptr_t)gsrc,     // global flat addr == AS(1) addr
      (las_v4i_ptr)lds_dst,             // generic -> AS(3) addrspacecast
      /*offset=*/0, /*cpol=*/CPOL_NT);  // stream-once: don't pollute L2
#else
  unsigned lds_addr = (unsigned)(uintptr_t)lds_dst;
  asm volatile("global_load_async_to_lds_b128 %0, %1, off th:TH_LOAD_NT"
               :: "v"(lds_addr), "v"(gsrc)
               : "memory");
#endif
}

template <int CNT>
__device__ __forceinline__ void wait_asynccnt() {
#if defined(__has_builtin) && __has_builtin(__builtin_amdgcn_s_wait_asynccnt)
  __builtin_amdgcn_s_wait_asynccnt(CNT);
#else
  asm volatile("s_wait_asynccnt %0" :: "i"(CNT) : "memory");
#endif
}

__launch_bounds__(TPB)
__global__ void router_fused_kernel(
    const float* __restrict__ sources,      // (P, N_SRC, DIM), P = B*S
    const float* __restrict__ w_all,        // (R, DIM)
    const float* __restrict__ mem_bias,     // (R,)
    const float* __restrict__ recent_bias,  // (R,)
    const float* __restrict__ norm_weight,  // (DIM,)
    const int*   __restrict__ router_idx_p, // scalar
    float* __restrict__ h_out,              // (P, DIM)
    float* __restrict__ alpha_out,          // (P,)
    int total_pos)
{
  __shared__ float stage[2][N_SRC * DIM];   // 2 x 24KB double buffer
  __shared__ float red[WAVES][2 * N_SRC];   // cross-wave reduction scratch

  const int t    = threadIdx.x;
  const int wave = t >> 5;
  const int lane = t & 31;
  const int d0   = t << 2;                  // this thread owns dims [d0, d0+3]

  const int    r  = __ldg(router_idx_p);
  const float4 wv = *(const float4*)(w_all + (size_t)r * DIM + d0);
  const float4 nw = *(const float4*)(norm_weight + d0);
  const float4 wg = make_float4(wv.x * nw.x, wv.y * nw.y, wv.z * nw.z, wv.w * nw.w);
  const float  mb = mem_bias[r];
  const float  rb = recent_bias[r];

  const int pos0 = blockIdx.x * PPB;
  if (pos0 >= total_pos) return;
  const int iters = min(PPB, total_pos - pos0);

  // Prime the pipeline: stage tile for position pos0 into buffer 0.
  {
    const float* src = sources + (size_t)pos0 * (N_SRC * DIM) + d0;
    float* dst = &stage[0][d0];
#pragma unroll
    for (int n = 0; n < N_SRC; ++n)
      async_f4_to_lds(src + n * DIM, dst + n * DIM);
  }

  for (int it = 0; it < iters; ++it) {
    const int cur = it & 1;
    const int pos = pos0 + it;

    if (it + 1 < iters) {
      // Prefetch next tile into the other buffer (its previous readers were
      // this thread's own ds_loads, already consumed => WAR safe, no barrier).
      const float* src = sources + (size_t)(pos + 1) * (N_SRC * DIM) + d0;
      float* dst = &stage[cur ^ 1][d0];
#pragma unroll
      for (int n = 0; n < N_SRC; ++n)
        async_f4_to_lds(src + n * DIM, dst + n * DIM);
      wait_asynccnt<N_SRC>();   // current tile done; next tile's 6 stay in flight
    } else {
      wait_asynccnt<0>();
    }

    // Read back own staged slots; per-thread partials.
    float4 xs[N_SRC];
    float  ps[2 * N_SRC];       // [0..5] = sum(x^2), [6..11] = dot(w*g, x)
#pragma unroll
    for (int n = 0; n < N_SRC; ++n) {
      xs[n] = *(const float4*)&stage[cur][n * DIM + d0];
      const float4 x = xs[n];
      ps[n]         = x.x * x.x + x.y * x.y + x.z * x.z + x.w * x.w;
      ps[N_SRC + n] = wg.x * x.x + wg.y * x.y + wg.z * x.z + wg.w * x.w;
    }

    // wave32 butterfly reduction (all lanes end with the wave total)
#pragma unroll
    for (int k = 0; k < 2 * N_SRC; ++k) {
#pragma unroll
      for (int m = 16; m >= 1; m >>= 1)
        ps[k] += __shfl_xor(ps[k], m, 32);
    }

    __syncthreads();            // protect red[] vs previous iteration's readers
    if (lane == 0) {
#pragma unroll
      for (int k = 0; k < 2 * N_SRC; ++k) red[wave][k] = ps[k];
    }
    __syncthreads();

    float ss[N_SRC], dw[N_SRC];
#pragma unroll
    for (int n = 0; n < N_SRC; ++n) { ss[n] = 0.f; dw[n] = 0.f; }
#pragma unroll
    for (int w = 0; w < WAVES; ++w) {
#pragma unroll
      for (int n = 0; n < N_SRC; ++n) {   // same-address reads broadcast in LDS
        ss[n] += red[w][n];
        dw[n] += red[w][N_SRC + n];
      }
    }

    // scores: dot(w*g, x) * rsqrt(mean(x^2) + eps), then biases
    float score[N_SRC];
#pragma unroll
    for (int n = 0; n < N_SRC; ++n)
      score[n] = dw[n] * (1.0f / sqrtf(ss[n] * (1.0f / DIM) + RMS_EPS));
    score[0]         += mb;
    score[N_SRC - 1] += rb;

    // softmax over N_SRC (redundant per thread; 6-wide, trivial)
    float mx = score[0];
#pragma unroll
    for (int n = 1; n < N_SRC; ++n) mx = fmaxf(mx, score[n]);
    float e[N_SRC], esum = 0.f;
#pragma unroll
    for (int n = 0; n < N_SRC; ++n) { e[n] = __expf(score[n] - mx); esum += e[n]; }
    const float inv = 1.0f / esum;

    // h = sum_n alpha_n * x_n  (raw sources), non-temporal b128 store
    v4f_t acc = {0.f, 0.f, 0.f, 0.f};
#pragma unroll
    for (int n = 0; n < N_SRC; ++n) {
      const float a = e[n] * inv;
      acc.x += a * xs[n].x; acc.y += a * xs[n].y;
      acc.z += a * xs[n].z; acc.w += a * xs[n].w;
    }
    __builtin_nontemporal_store(acc, (v4f_t*)(h_out + (size_t)pos * DIM + d0));
    if (t == 0) __builtin_nontemporal_store(e[0] * inv, alpha_out + pos);
  }
}

extern "C" void kernel_launch(void* const* d_in, const int* in_sizes, int n_in,
                              void* d_out, int out_size, void* d_ws, size_t ws_size,
                              hipStream_t stream) {
  const float* sources     = (const float*)d_in[0];
  const float* w_all       = (const float*)d_in[1];
  const float* mem_bias    = (const float*)d_in[2];
  const float* recent_bias = (const float*)d_in[3];
  const float* norm_weight = (const float*)d_in[4];
  const int*   router_idx  = (const int*)d_in[5];
  float* out = (float*)d_out;

  const int d         = in_sizes[4];                 // 1024
  const int total_pos = in_sizes[0] / (N_SRC * d);   // B*S = 16384

  float* h_out     = out;
  float* alpha_out = out + (size_t)total_pos * d;

  const int blocks = (total_pos + PPB - 1) / PPB;
  hipLaunchKernelGGL(router_fused_kernel, dim3(blocks), dim3(TPB), 0, stream,
                     sources, w_all, mem_bias, recent_bias, norm_weight,
                     router_idx, h_out, alpha_out, total_pos);
}